// HexEye_25795573579987
// MI455X (gfx1250) — compile-verified
//
#include <hip/hip_runtime.h>
#include <hip/hip_bf16.h>

typedef __attribute__((ext_vector_type(2))) float v2f;
typedef __attribute__((ext_vector_type(8))) float v8f;

#define H_PX   600
#define W_PX   800
#define N_OMM  721
#define PADR   4
#define KK     81          // 9x9 patch
#define NTILES 46          // ceil(721/16)
#define NPLANES 96         // 32 batches * 3 channels

__device__ __forceinline__ int reflect_idx(int v, int n) {
    v = (v < 0) ? -v : v;                 // jnp.pad 'reflect' (no edge repeat); PAD=4 -> one bounce
    v = (v >= n) ? (2 * n - 2 - v) : v;
    return v;
}

__global__ __launch_bounds__(256) void hexeye_wmma_f32(
    const float* __restrict__ stim,   // (96, 600, 800) flattened planes
    const int*   __restrict__ rx,     // (721)
    const int*   __restrict__ ry,     // (721)
    float*       __restrict__ out)    // (96, 721)
{
    const int wavesPerBlock = blockDim.x >> 5;
    const int wave = blockIdx.x * wavesPerBlock + (threadIdx.x >> 5);
    const int lane = threadIdx.x & 31;

    const int plane = wave / NTILES;          // b*3 + c  (wave-uniform)
    const int tile  = wave - plane * NTILES;
    if (plane >= NPLANES) return;             // whole-wave exit

    const int n   = lane & 15;                // receptor column within tile (N of WMMA)
    const int sub = lane >> 4;                // which K rows this lane feeds (sub, sub+2)
    const int o_raw = tile * 16 + n;
    const bool valid = (o_raw < N_OMM);
    const int o = valid ? o_raw : (N_OMM - 1);

    int cx = rx[o]; cx = min(max(cx, PADR), W_PX + PADR - 1);
    int cy = ry[o]; cy = min(max(cy, PADR), H_PX + PADR - 1);

    const float* __restrict__ pp = stim + (size_t)plane * (H_PX * W_PX);

    // Hoisted reflection: 9 row bases + 9 columns per lane, all later indices compile-time.
    int Yb[9], Xc[9];
    #pragma unroll
    for (int d = 0; d < 9; ++d) {
        Yb[d] = reflect_idx(cy + d - 8, H_PX) * W_PX;
        Xc[d] = reflect_idx(cx + d - 8, W_PX);
    }

    // A = 16x4 all-ones (exact *1.0 multiplies -> pure f32 accumulation)
    v2f a; a.x = 1.0f; a.y = 1.0f;
    v8f acc = {};

    // Fully unrolled: step s consumes flat patch elements 4s..4s+3 across K rows 0..3.
    // Lane with `sub` supplies rows sub (B vgpr0) and sub+2 (B vgpr1).
    #pragma unroll
    for (int s = 0; s < 21; ++s) {
        float b0, b1;
        if (s < 20) {
            const int k0a = 4 * s + 0, k0b = 4 * s + 1;   // row 0 / row 1 elements
            const int k1a = 4 * s + 2, k1b = 4 * s + 3;   // row 2 / row 3 elements
            // compile-time /9 and %9 -> one cndmask per address
            const int i0 = sub ? (Yb[k0b / 9] + Xc[k0b % 9])
                               : (Yb[k0a / 9] + Xc[k0a % 9]);
            const int i1 = sub ? (Yb[k1b / 9] + Xc[k1b % 9])
                               : (Yb[k1a / 9] + Xc[k1a % 9]);
            b0 = pp[i0];
            b1 = pp[i1];
        } else {
            // tail: only element 80 (oy=8, ox=8) is real; rows 1..3 are zero padding
            const float v = pp[Yb[8] + Xc[8]];            // all lanes load a safe address
            b0 = sub ? 0.0f : v;                          // branchless mask, EXEC stays all-1s
            b1 = 0.0f;
        }
        v2f bb; bb.x = b0; bb.y = b1;
        // D = ones(16x4) x B(4x16) + C  -> every D row = column sums of B
        acc = __builtin_amdgcn_wmma_f32_16x16x4_f32(
            /*neg_a=*/false, a, /*neg_b=*/false, bb,
            /*c_mod=*/(short)0, acc, /*reuse_a=*/false, /*reuse_b=*/false);
    }

    // D layout (f32 16x16): acc[0] on lanes 0..15 is D[M=0][N=lane] = sum for receptor tile*16+lane
    if (valid && sub == 0) {
        out[(size_t)plane * N_OMM + o_raw] = acc[0] / 81.0f;
    }
}

extern "C" void kernel_launch(void* const* d_in, const int* in_sizes, int n_in,
                              void* d_out, int out_size, void* d_ws, size_t ws_size,
                              hipStream_t stream) {
    const float* stim = (const float*)d_in[0];   // (32,3,600,800) f32
    const int*   rx   = (const int*)d_in[1];     // (721) i32
    const int*   ry   = (const int*)d_in[2];     // (721) i32
    float*       out  = (float*)d_out;           // (32,3,721) f32

    (void)in_sizes; (void)n_in; (void)out_size; (void)d_ws; (void)ws_size;

    const int totalWaves = NPLANES * NTILES;     // 4416 wave32s
    const int wavesPerBlock = 256 / 32;          // 8
    const int blocks = (totalWaves + wavesPerBlock - 1) / wavesPerBlock;  // 552
    hexeye_wmma_f32<<<blocks, 256, 0, stream>>>(stim, rx, ry, out);
}